// GraphOne_86019605004433
// MI455X (gfx1250) — compile-verified
//
#include <hip/hip_runtime.h>

#define NN 100000
#define EE 1600000
#define ET (EE + NN)
#define FF 128
#define GG 64

typedef __attribute__((ext_vector_type(2))) float v2f;
typedef __attribute__((ext_vector_type(8))) float v8f;

// ---------- helpers ----------
__device__ inline unsigned fkey(float f) {
    unsigned b = __float_as_uint(f);
    return (b & 0x80000000u) ? ~b : (b | 0x80000000u);
}
__device__ inline float fval(unsigned k) {
    return (k & 0x80000000u) ? __uint_as_float(k & 0x7fffffffu) : __uint_as_float(~k);
}
__device__ inline void edge_sd(const long long* __restrict__ ei, int e, int& s, int& d) {
    if (e < EE) { s = (int)ei[e]; d = (int)ei[EE + e]; }
    else        { s = e - EE;     d = e - EE; }
}

// ---------- GEMM: C[NN x FF] = A[NN x FF] @ W[FF x FF], f32 WMMA ----------
__global__ __launch_bounds__(256) void gemm_wmma_kernel(const float* __restrict__ A,
                                                        const float* __restrict__ W,
                                                        float* __restrict__ C) {
    __shared__ float sW[FF * FF];              // 64 KB: whole weight matrix in LDS
    int tid = threadIdx.x;
    const float4* Wv = (const float4*)W;
    float4* sWv = (float4*)sW;
#pragma unroll
    for (int i = 0; i < (FF * FF / 4) / 256; ++i)
        sWv[tid + i * 256] = Wv[tid + i * 256];
    __syncthreads();

    int lane = tid & 31;
    int wave = tid >> 5;
    int rowBase = blockIdx.x * 16;             // NN = 6250 * 16, exact
    int colBase = wave * 16;                   // 8 waves cover 128 cols
    int m = lane & 15;
    int koff = (lane >> 4) * 2;                // lanes 0-15: K+0/1, lanes 16-31: K+2/3

    v8f acc = {};
    const float* arow = A + (size_t)(rowBase + m) * FF;
#pragma unroll
    for (int k = 0; k < FF; k += 4) {
        v2f a = *(const v2f*)(arow + k + koff);
        v2f b;
        b.x = sW[(k + koff) * FF + colBase + m];
        b.y = sW[(k + koff + 1) * FF + colBase + m];
        acc = __builtin_amdgcn_wmma_f32_16x16x4_f32(false, a, false, b,
                                                    (short)0, acc, false, false);
    }
    int ro = (lane & 16) ? 8 : 0;
#pragma unroll
    for (int i = 0; i < 8; ++i)
        C[(size_t)(rowBase + i + ro) * FF + colBase + m] = acc[i];
}

// ---------- alpha_src / alpha_dst: per-node dot products (one wave per node) ----------
__global__ __launch_bounds__(256) void alphas_kernel(const float* __restrict__ h,
                                                     const float* __restrict__ avs,
                                                     const float* __restrict__ avd,
                                                     float* __restrict__ as,
                                                     float* __restrict__ ad) {
    int lane = threadIdx.x & 31, wave = threadIdx.x >> 5;
    int n = blockIdx.x * 8 + wave;
    if (n >= NN) return;
    float4 hv = *(const float4*)(h + (size_t)n * FF + lane * 4);
    float4 s4 = *(const float4*)(avs + lane * 4);
    float4 d4 = *(const float4*)(avd + lane * 4);
    float ps = hv.x * s4.x + hv.y * s4.y + hv.z * s4.z + hv.w * s4.w;
    float pd = hv.x * d4.x + hv.y * d4.y + hv.z * d4.z + hv.w * d4.w;
    for (int off = 16; off > 0; off >>= 1) {
        ps += __shfl_down(ps, off, 32);
        pd += __shfl_down(pd, off, 32);
    }
    if (lane == 0) { as[n] = ps; ad[n] = pd; }
}

// ---------- segment softmax over dst: max pass ----------
__global__ void edge_max_kernel(const long long* __restrict__ ei,
                                const float* __restrict__ as,
                                const float* __restrict__ ad,
                                unsigned* __restrict__ mkey) {
    int e = blockIdx.x * 256 + threadIdx.x;
    if (e >= ET) return;
    int s, d; edge_sd(ei, e, s, d);
    float v = as[s] + ad[d];
    v = v > 0.f ? v : 0.2f * v;               // leaky_relu(0.2)
    atomicMax(&mkey[d], fkey(v));
}

// ---------- segment softmax: exp + denom pass (caches per-edge exp) ----------
__global__ void edge_exp_kernel(const long long* __restrict__ ei,
                                const float* __restrict__ as,
                                const float* __restrict__ ad,
                                const unsigned* __restrict__ mkey,
                                float* __restrict__ denom,
                                float* __restrict__ eexp) {
    int e = blockIdx.x * 256 + threadIdx.x;
    if (e >= ET) return;
    int s, d; edge_sd(ei, e, s, d);
    float v = as[s] + ad[d];
    v = v > 0.f ? v : 0.2f * v;
    float ex = __expf(v - fval(mkey[d]));
    eexp[e] = ex;
    atomicAdd(&denom[d], ex);
}

// ---------- weighted scatter: out[dst] += coef * h[src] (wave per edge) ----------
__global__ __launch_bounds__(256) void edge_scatter_kernel(const long long* __restrict__ ei,
                                                           const float* __restrict__ eexp,
                                                           const float* __restrict__ denom,
                                                           const float* __restrict__ hs,
                                                           float* __restrict__ out) {
    int lane = threadIdx.x & 31, wave = threadIdx.x >> 5;
    int e = blockIdx.x * 8 + wave;
    if (e >= ET) return;
    int s, d; edge_sd(ei, e, s, d);
    float coef = eexp[e] / denom[d];
    float4 hv = *(const float4*)(hs + (size_t)s * FF + lane * 4);
    float* o = out + (size_t)d * FF + lane * 4;
    atomicAdd(o + 0, coef * hv.x);
    atomicAdd(o + 1, coef * hv.y);
    atomicAdd(o + 2, coef * hv.z);
    atomicAdd(o + 3, coef * hv.w);
}

// ---------- bias + leaky_relu(0.01), in place ----------
__global__ void bias_lrelu_kernel(float* __restrict__ h, const float* __restrict__ b) {
    size_t i = (size_t)blockIdx.x * 256 + threadIdx.x;
    if (i >= (size_t)NN * FF) return;
    float v = h[i] + b[i & (FF - 1)];
    h[i] = v > 0.f ? v : 0.01f * v;
}

// ---------- gate = h @ gate_W + gate_b; per-graph max (wave per node) ----------
__global__ __launch_bounds__(256) void gate_kernel(const float* __restrict__ h,
                                                   const float* __restrict__ gW,
                                                   const float* __restrict__ gb,
                                                   const long long* __restrict__ batch,
                                                   float* __restrict__ gate,
                                                   unsigned* __restrict__ gkey) {
    int lane = threadIdx.x & 31, wave = threadIdx.x >> 5;
    int n = blockIdx.x * 8 + wave;
    if (n >= NN) return;
    float4 hv = *(const float4*)(h + (size_t)n * FF + lane * 4);
    float4 w4 = *(const float4*)(gW + lane * 4);
    float p = hv.x * w4.x + hv.y * w4.y + hv.z * w4.z + hv.w * w4.w;
    for (int off = 16; off > 0; off >>= 1) p += __shfl_down(p, off, 32);
    if (lane == 0) {
        p += gb[0];
        gate[n] = p;
        atomicMax(&gkey[(int)batch[n]], fkey(p));
    }
}

__global__ void gate_exp_kernel(const float* __restrict__ gate,
                                const long long* __restrict__ batch,
                                const unsigned* __restrict__ gkey,
                                float* __restrict__ gdenom,
                                float* __restrict__ wnode) {
    int n = blockIdx.x * 256 + threadIdx.x;
    if (n >= NN) return;
    int g = (int)batch[n];
    float ex = __expf(gate[n] - fval(gkey[g]));
    wnode[n] = ex;
    atomicAdd(&gdenom[g], ex);
}

// ---------- pooled[g] += w * h[n] (wave per node) ----------
__global__ __launch_bounds__(256) void pool_kernel(const float* __restrict__ h,
                                                   const float* __restrict__ wnode,
                                                   const float* __restrict__ gdenom,
                                                   const long long* __restrict__ batch,
                                                   float* __restrict__ pooled) {
    int lane = threadIdx.x & 31, wave = threadIdx.x >> 5;
    int n = blockIdx.x * 8 + wave;
    if (n >= NN) return;
    int g = (int)batch[n];
    float w = wnode[n] / gdenom[g];
    float4 hv = *(const float4*)(h + (size_t)n * FF + lane * 4);
    float* o = pooled + (size_t)g * FF + lane * 4;
    atomicAdd(o + 0, w * hv.x);
    atomicAdd(o + 1, w * hv.y);
    atomicAdd(o + 2, w * hv.z);
    atomicAdd(o + 3, w * hv.w);
}

// ---------- out[g] = pooled[g] @ fin_W + fin_b (wave per graph) ----------
__global__ __launch_bounds__(256) void final_kernel(const float* __restrict__ pooled,
                                                    const float* __restrict__ fW,
                                                    const float* __restrict__ fb,
                                                    float* __restrict__ out) {
    int lane = threadIdx.x & 31, wave = threadIdx.x >> 5;
    int g = blockIdx.x * 8 + wave;
    if (g >= GG) return;
    float4 pv = *(const float4*)(pooled + (size_t)g * FF + lane * 4);
    float4 w4 = *(const float4*)(fW + lane * 4);
    float p = pv.x * w4.x + pv.y * w4.y + pv.z * w4.z + pv.w * w4.w;
    for (int off = 16; off > 0; off >>= 1) p += __shfl_down(p, off, 32);
    if (lane == 0) out[g] = p + fb[0];
}

extern "C" void kernel_launch(void* const* d_in, const int* in_sizes, int n_in,
                              void* d_out, int out_size, void* d_ws, size_t ws_size,
                              hipStream_t stream) {
    (void)in_sizes; (void)n_in; (void)out_size; (void)ws_size;
    const float*     x     = (const float*)d_in[0];
    const long long* ei    = (const long long*)d_in[1];
    const long long* batch = (const long long*)d_in[2];
    const float* W1  = (const float*)d_in[3];
    const float* as1 = (const float*)d_in[4];
    const float* ad1 = (const float*)d_in[5];
    const float* b1  = (const float*)d_in[6];
    const float* W2  = (const float*)d_in[7];
    const float* as2 = (const float*)d_in[8];
    const float* ad2 = (const float*)d_in[9];
    const float* b2  = (const float*)d_in[10];
    const float* gW  = (const float*)d_in[11];
    const float* gb  = (const float*)d_in[12];
    const float* fW  = (const float*)d_in[13];
    const float* fb  = (const float*)d_in[14];
    float* out = (float*)d_out;

    char* wsp = (char*)d_ws;
    auto alloc = [&](size_t bytes) -> char* {
        char* p = wsp;
        wsp += (bytes + 255) & ~(size_t)255;
        return p;
    };
    float*    h1     = (float*)alloc((size_t)NN * FF * 4);
    float*    h2     = (float*)alloc((size_t)NN * FF * 4);
    float*    as     = (float*)alloc((size_t)NN * 4);
    float*    ad     = (float*)alloc((size_t)NN * 4);
    unsigned* mkey   = (unsigned*)alloc((size_t)NN * 4);
    float*    denom  = (float*)alloc((size_t)NN * 4);
    float*    eexp   = (float*)alloc((size_t)ET * 4);
    float*    gate   = (float*)alloc((size_t)NN * 4);
    float*    wnode  = (float*)alloc((size_t)NN * 4);
    unsigned* gkey   = (unsigned*)alloc((size_t)GG * 4);
    float*    gdenom = (float*)alloc((size_t)GG * 4);
    float*    pooled = (float*)alloc((size_t)GG * FF * 4);

    const int gemm_grid   = NN / 16;            // 6250
    const int wavepn_grid = (NN + 7) / 8;       // 12500
    const int node_grid   = (NN + 255) / 256;
    const int edge_grid   = (ET + 255) / 256;
    const int escat_grid  = (ET + 7) / 8;
    const int nf_grid     = (int)(((size_t)NN * FF + 255) / 256);

    auto layer = [&](const float* hin, const float* W, const float* avs,
                     const float* avd, const float* bias) {
        gemm_wmma_kernel<<<gemm_grid, 256, 0, stream>>>(hin, W, h1);
        alphas_kernel<<<wavepn_grid, 256, 0, stream>>>(h1, avs, avd, as, ad);
        hipMemsetAsync(mkey, 0, (size_t)NN * 4, stream);
        hipMemsetAsync(denom, 0, (size_t)NN * 4, stream);
        edge_max_kernel<<<edge_grid, 256, 0, stream>>>(ei, as, ad, mkey);
        edge_exp_kernel<<<edge_grid, 256, 0, stream>>>(ei, as, ad, mkey, denom, eexp);
        hipMemsetAsync(h2, 0, (size_t)NN * FF * 4, stream);
        edge_scatter_kernel<<<escat_grid, 256, 0, stream>>>(ei, eexp, denom, h1, h2);
        bias_lrelu_kernel<<<nf_grid, 256, 0, stream>>>(h2, bias);
    };

    layer(x, W1, as1, ad1, b1);   // h2 = leaky_relu(gat1(x))
    layer(h2, W2, as2, ad2, b2);  // h2 = leaky_relu(gat2(h2))  (gemm reads h2 before memset)

    hipMemsetAsync(gkey, 0, (size_t)GG * 4, stream);
    hipMemsetAsync(gdenom, 0, (size_t)GG * 4, stream);
    hipMemsetAsync(pooled, 0, (size_t)GG * FF * 4, stream);
    gate_kernel<<<wavepn_grid, 256, 0, stream>>>(h2, gW, gb, batch, gate, gkey);
    gate_exp_kernel<<<node_grid, 256, 0, stream>>>(gate, batch, gkey, gdenom, wnode);
    pool_kernel<<<wavepn_grid, 256, 0, stream>>>(h2, wnode, gdenom, batch, pooled);
    final_kernel<<<(GG + 7) / 8, 256, 0, stream>>>(pooled, fW, fb, out);
}